// GDP_1305670058177
// MI455X (gfx1250) — compile-verified
//
#include <hip/hip_runtime.h>
#include <stdint.h>

typedef __attribute__((ext_vector_type(4))) float v4f;
typedef __attribute__((ext_vector_type(4))) int   v4i;

#define C_DIM 64
#define H_DIM 480
#define W_DIM 640
#define HW_DIM (H_DIM * W_DIM)

// ---------------------------------------------------------------------------
// Phase 1: transpose x2d [C][H*W] -> xT [H*W][C].
// Tile load uses gfx1250 GLOBAL_LOAD_ASYNC_TO_LDS_B128 (ASYNCcnt-tracked,
// bypasses VGPRs); write-out uses b128 stores. 8 VMEM ops per thread total.
// LDS tile padded to 68 floats: keeps every b128 16B-aligned, avoids bank
// conflicts on the transposed read.
// ---------------------------------------------------------------------------
__global__ __launch_bounds__(256) void gdp_xpose_kernel(
    const float* __restrict__ x2d, float* __restrict__ xT)
{
    __shared__ float tile[64][68];
    const int p0 = blockIdx.x * 64;         // pixel tile base
    const int w  = threadIdx.x;             // 0..255
    const int pg = w & 15;                  // pixel group: 4 consecutive pixels
    const int c0 = w >> 4;                  // 0..15

#pragma unroll
    for (int k = 0; k < 4; ++k) {
        const int c = c0 + 16 * k;
        const float* gp = x2d + (size_t)c * HW_DIM + p0 + pg * 4;
        const uint32_t lds_off = (uint32_t)(uintptr_t)&tile[c][pg * 4];
        asm volatile("global_load_async_to_lds_b128 %0, %1, off"
                     :: "v"(lds_off), "v"((uint64_t)(uintptr_t)gp)
                     : "memory");
    }
    asm volatile("s_wait_asynccnt 0" ::: "memory");
    __syncthreads();

#pragma unroll
    for (int k = 0; k < 4; ++k) {
        const int p  = (w >> 4) + 16 * k;   // 0..63
        const int cg = (w & 15) * 4;        // channel group of 4
        const v4f o = { tile[cg + 0][p], tile[cg + 1][p],
                        tile[cg + 2][p], tile[cg + 3][p] };
        *(v4f*)(xT + (size_t)(p0 + p) * C_DIM + cg) = o;   // coalesced b128
    }
}

// ---------------------------------------------------------------------------
// Phase 2: gather + Gaussian weight + channel-major streaming store.
// Each thread owns 4 consecutive voxels: 16x (4 b128 gathers -> 4x4 register
// transpose -> 4 b128 NON-TEMPORAL stores). Output (512MB) streams past L2 so
// xT (78MB) + depth stay resident in the 192MB L2.
// ---------------------------------------------------------------------------
__global__ __launch_bounds__(256) void gdp_project_T_kernel(
    const float*   __restrict__ xT,       // [H*W][64]
    const int*     __restrict__ ppix,     // [N][2]
    const int*     __restrict__ scale_p,  // scalar
    const uint8_t* __restrict__ fov,      // [N] bool bytes
    const float*   __restrict__ pix_z,    // [N]
    const float*   __restrict__ depth,    // [H*W]
    float*         __restrict__ out,      // [C][N]
    int N)
{
    const int t  = blockIdx.x * blockDim.x + threadIdx.x;
    const int n0 = t * 4;
    if (n0 >= N) return;

    const int scale = scale_p[0];

    const v4i pa = ((const v4i*)ppix)[2 * t + 0];   // x0 y0 x1 y1
    const v4i pb = ((const v4i*)ppix)[2 * t + 1];   // x2 y2 x3 y3
    const int px[4] = { pa.x, pa.z, pb.x, pb.z };
    const int py[4] = { pa.y, pa.w, pb.y, pb.w };

    uint32_t m4;
    __builtin_memcpy(&m4, fov + n0, 4);             // 4 bool bytes
    const v4f z4 = ((const v4f*)pix_z)[t];
    const float zz[4] = { z4.x, z4.y, z4.z, z4.w };

    const v4f* rows[4];
    float wgt[4];
#pragma unroll
    for (int i = 0; i < 4; ++i) {
        int fx = px[i], fy = py[i];
        if (scale != 1) { fx = px[i] / scale; fy = py[i] / scale; }
        const size_t fidx = (size_t)(fy * W_DIM + fx);
        const float* rp = xT + fidx * C_DIM;        // 256B-aligned row
        rows[i] = (const v4f*)rp;
        __builtin_prefetch(rp);                      // global_prefetch_b8
        __builtin_prefetch(rp + 32);                 // second 128B line
        const bool  m  = ((m4 >> (8 * i)) & 0xffu) != 0u;
        const float dv = m ? depth[py[i] * W_DIM + px[i]] : 0.0f;
        const float d  = zz[i] - dv;
        const float gw = __expf(-0.5f * d * d);      // v_exp_f32
        const float wv = (dv == 0.0f) ? 1.0f : gw;
        wgt[i] = m ? wv : 0.0f;
    }

    float* op = out + n0;                            // running channel pointer
#pragma unroll 4
    for (int cb = 0; cb < 16; ++cb) {                // 4 channels per block
        const v4f v0 = rows[0][cb] * wgt[0];
        const v4f v1 = rows[1][cb] * wgt[1];
        const v4f v2 = rows[2][cb] * wgt[2];
        const v4f v3 = rows[3][cb] * wgt[3];
        // 4x4 register transpose -> channel-major NT b128 stores.
        const v4f o0 = { v0.x, v1.x, v2.x, v3.x };
        const v4f o1 = { v0.y, v1.y, v2.y, v3.y };
        const v4f o2 = { v0.z, v1.z, v2.z, v3.z };
        const v4f o3 = { v0.w, v1.w, v2.w, v3.w };
        __builtin_nontemporal_store(o0, (v4f*)op); op += N;
        __builtin_nontemporal_store(o1, (v4f*)op); op += N;
        __builtin_nontemporal_store(o2, (v4f*)op); op += N;
        __builtin_nontemporal_store(o3, (v4f*)op); op += N;
    }
}

// ---------------------------------------------------------------------------
// Fallback (workspace too small): direct gather from channel-major x2d.
// ---------------------------------------------------------------------------
__global__ __launch_bounds__(256) void gdp_project_direct_kernel(
    const float*   __restrict__ x2d,
    const int*     __restrict__ ppix,
    const int*     __restrict__ scale_p,
    const uint8_t* __restrict__ fov,
    const float*   __restrict__ pix_z,
    const float*   __restrict__ depth,
    float*         __restrict__ out,
    int N)
{
    const int t  = blockIdx.x * blockDim.x + threadIdx.x;
    const int n0 = t * 4;
    if (n0 >= N) return;

    const int scale = scale_p[0];
    const v4i pa = ((const v4i*)ppix)[2 * t + 0];
    const v4i pb = ((const v4i*)ppix)[2 * t + 1];
    const int px[4] = { pa.x, pa.z, pb.x, pb.z };
    const int py[4] = { pa.y, pa.w, pb.y, pb.w };

    uint32_t m4;
    __builtin_memcpy(&m4, fov + n0, 4);
    const v4f z4 = ((const v4f*)pix_z)[t];
    const float zz[4] = { z4.x, z4.y, z4.z, z4.w };

    size_t fidx[4];
    float  wgt[4];
#pragma unroll
    for (int i = 0; i < 4; ++i) {
        int fx = px[i], fy = py[i];
        if (scale != 1) { fx = px[i] / scale; fy = py[i] / scale; }
        fidx[i] = (size_t)(fy * W_DIM + fx);
        const bool  m  = ((m4 >> (8 * i)) & 0xffu) != 0u;
        const float dv = m ? depth[py[i] * W_DIM + px[i]] : 0.0f;
        const float d  = zz[i] - dv;
        const float gw = __expf(-0.5f * d * d);
        const float wv = (dv == 0.0f) ? 1.0f : gw;
        wgt[i] = m ? wv : 0.0f;
    }

    float* op = out + n0;
#pragma unroll 4
    for (int c = 0; c < C_DIM; ++c) {
        const float* ch = x2d + (size_t)c * HW_DIM;
        const v4f o = { ch[fidx[0]] * wgt[0], ch[fidx[1]] * wgt[1],
                        ch[fidx[2]] * wgt[2], ch[fidx[3]] * wgt[3] };
        __builtin_nontemporal_store(o, (v4f*)op);
        op += N;
    }
}

extern "C" void kernel_launch(void* const* d_in, const int* in_sizes, int n_in,
                              void* d_out, int out_size, void* d_ws, size_t ws_size,
                              hipStream_t stream)
{
    const float*   x2d   = (const float*)d_in[0];
    const int*     ppix  = (const int*)d_in[1];
    const int*     scale = (const int*)d_in[2];
    const uint8_t* fov   = (const uint8_t*)d_in[3];   // jnp.bool_ -> 1 byte/elt
    const float*   pix_z = (const float*)d_in[4];
    const float*   depth = (const float*)d_in[5];
    float*         out   = (float*)d_out;

    const int N = in_sizes[4];                         // pix_z has N elements
    const int nThreads = (N + 3) / 4;
    const int nBlocks  = (nThreads + 255) / 256;

    const size_t xT_bytes = (size_t)C_DIM * HW_DIM * sizeof(float); // 78.6 MB
    if (ws_size >= xT_bytes) {
        float* xT = (float*)d_ws;
        gdp_xpose_kernel<<<HW_DIM / 64, 256, 0, stream>>>(x2d, xT);
        gdp_project_T_kernel<<<nBlocks, 256, 0, stream>>>(
            xT, ppix, scale, fov, pix_z, depth, out, N);
    } else {
        gdp_project_direct_kernel<<<nBlocks, 256, 0, stream>>>(
            x2d, ppix, scale, fov, pix_z, depth, out, N);
    }
}